// MeshDownConv_37623913513298
// MI455X (gfx1250) — compile-verified
//
#include <hip/hip_runtime.h>

// ---------------------------------------------------------------------------
// SplineConv (quadratic B-spline, 3x3 taps) x2 on MI455X / gfx1250.
//
// Restructured as: T = X @ [W0..W8 | root]  (f32 WMMA GEMM, [N,64]@[64,640])
// then per-edge   msg_e = sum_k B[e,k] * T[src_e, k*64 : k*64+64]
// scatter-added with f32 global atomics into out[dst_e]; out preinitialized
// with the root/bias term; ReLU at the end. 14x fewer FLOPs than reference;
// runtime bound by the contiguous 2304B/edge gather of T (~3.7GB/layer,
// ~0.16ms at 23.3 TB/s), not by the 8.2 GFLOP/layer GEMM.
//
// LDS staging stores K-adjacent value PAIRS (float2) so every WMMA fragment
// is a single aligned ds_load_b64 -> no VGPR-pair assembly movs.
// ---------------------------------------------------------------------------

typedef float v2f __attribute__((ext_vector_type(2)));
typedef float v8f __attribute__((ext_vector_type(8)));

#define NNODES 100000
#define NEDGES 1600000
#define CIN 64
#define COUT 64
#define NUMK 9
#define LDW 640            // 9*64 message taps + 64 root columns
#define AP_STRIDE 17       // float2 units; write banks 2*(17p mod 32): conflict-free
#define BP_STRIDE 144      // float2 units; 288 floats == 32 mod 64 banks:
                           // half-wave read windows exactly disjoint

// Repack W[9,64,64] (k, cin, cout) and root[64,64] into Wf[64, 640]:
// Wf[cin, k*64+cout] = W[k,cin,cout]; Wf[cin, 576+cout] = root[cin,cout].
__global__ __launch_bounds__(256) void repack_kernel(const float* __restrict__ W,
                                                     const float* __restrict__ root,
                                                     float* __restrict__ Wf) {
    int t = blockIdx.x * 256 + threadIdx.x;
    if (t >= CIN * LDW) return;
    int cin = t / LDW;
    int j   = t - cin * LDW;
    float v;
    if (j < NUMK * COUT) {
        int k = j >> 6, cout = j & 63;
        v = W[((size_t)k * CIN + cin) * COUT + cout];
    } else {
        v = root[(size_t)cin * COUT + (j - NUMK * COUT)];
    }
    Wf[t] = v;
}

// T[rb:rb+16, cb:cb+128] = X[rb:rb+16, 0:64] @ Wf[0:64, cb:cb+128]
// 8 waves per block; each wave owns a 16x16 tile via v_wmma_f32_16x16x4_f32.
__global__ __launch_bounds__(256) void spline_gemm_kernel(const float* __restrict__ X,
                                                          const float* __restrict__ Wf,
                                                          float* __restrict__ T,
                                                          int nrows) {
    // K-pair-major staging: index p = k/2 (0..31) selects an adjacent K pair.
    __shared__ float2 Ap2[32 * AP_STRIDE];   // Ap2[p][m]     = (X[m][2p], X[m][2p+1])
    __shared__ float2 Bp2[32 * BP_STRIDE];   // Bp2[p][n]     = (Wf[2p][n], Wf[2p+1][n])

    const int tid  = threadIdx.x;
    const int wave = tid >> 5;
    const int lane = tid & 31;
    const int rb   = blockIdx.x * 16;
    const int cb   = blockIdx.y * 128;

    // A tile 16x64: per thread one float2 from global (256B/wave coalesced),
    // stored pre-paired. 16 rows x 32 pairs / 256 threads = 2 iters.
#pragma unroll
    for (int i = 0; i < 2; ++i) {
        int mm = (tid >> 5) + i * 8;
        int pp = tid & 31;
        float2 v = make_float2(0.0f, 0.0f);
        if (rb + mm < nrows)
            v = *(const float2*)(X + (size_t)(rb + mm) * CIN + 2 * pp);
        Ap2[pp * AP_STRIDE + mm] = v;
    }
    // B tile 64x128 as 32 pair-rows: two float4 global reads (512B/wave rows),
    // interleaved into float2 pairs. 32 pair-rows x 128 cols / 256 thr = 4 iters.
#pragma unroll
    for (int i = 0; i < 4; ++i) {
        int p = (tid >> 5) + i * 8;
        int c = (tid & 31) * 4;
        float4 r0 = *(const float4*)(Wf + (size_t)(2 * p)     * LDW + cb + c);
        float4 r1 = *(const float4*)(Wf + (size_t)(2 * p + 1) * LDW + cb + c);
        float2* bp = &Bp2[p * BP_STRIDE + c];
        bp[0] = make_float2(r0.x, r1.x);
        bp[1] = make_float2(r0.y, r1.y);
        bp[2] = make_float2(r0.z, r1.z);
        bp[3] = make_float2(r0.w, r1.w);
    }
    __syncthreads();

    const int m    = lane & 15;
    const int half = lane >> 4;          // lanes 16-31 carry the K+2,K+3 pair
    const int nb   = wave * 16;

    v8f acc = {};
#pragma unroll
    for (int kk = 0; kk < 16; ++kk) {
        const int p = kk * 2 + half;
        v2f a = *(const v2f*)&Ap2[p * AP_STRIDE + m];
        v2f b = *(const v2f*)&Bp2[p * BP_STRIDE + nb + m];
        // (neg_a, A, neg_b, B, c_mod, C, reuse_a, reuse_b)
        acc = __builtin_amdgcn_wmma_f32_16x16x4_f32(false, a, false, b,
                                                    (short)0, acc, false, false);
    }

    // C/D layout: VGPR v -> M = v (lanes 0-15) / v+8 (lanes 16-31), N = lane&15
    const int row0 = rb + (half << 3);
    const int col  = cb + nb + m;
#pragma unroll
    for (int v = 0; v < 8; ++v) {
        int row = row0 + v;
        if (row < nrows) T[(size_t)row * LDW + col] = acc[v];
    }
}

// out[n,c] = T[n, 576+c] + bias[c]   (root term + bias; fully overwrites out)
__global__ __launch_bounds__(256) void init_out_kernel(const float* __restrict__ T,
                                                       const float* __restrict__ bias,
                                                       float* __restrict__ out) {
    int t = blockIdx.x * 256 + threadIdx.x;
    if (t >= NNODES * COUT) return;
    int n = t >> 6, c = t & 63;
    out[t] = T[(size_t)n * LDW + NUMK * COUT + c] + bias[c];
}

// One wave per edge, 2 channels per lane. Contiguous 2304B gather of T[src],
// 9 FMAs per channel, 2 f32 global atomics per lane into out[dst] (25.6MB
// accumulator -> L2-resident atomics).
__global__ __launch_bounds__(256) void edge_scatter_kernel(const int* __restrict__ src,
                                                           const int* __restrict__ dst,
                                                           const float* __restrict__ pseudo,
                                                           const float* __restrict__ T,
                                                           float* __restrict__ out) {
    int e = blockIdx.x * 8 + (threadIdx.x >> 5);
    if (e >= NEDGES) return;
    const int lane = threadIdx.x & 31;
    const int s = src[e];
    const int d = dst[e];
    const float u = pseudo[2 * e + 0];
    const float w = pseudo[2 * e + 1];

    float b0[3], b1[3];
    b0[0] = 0.5f * u * u - u + 0.5f;  b0[1] = -u * u + u + 0.5f;  b0[2] = 0.5f * u * u;
    b1[0] = 0.5f * w * w - w + 0.5f;  b1[1] = -w * w + w + 0.5f;  b1[2] = 0.5f * w * w;

    const float* Trow = T + (size_t)s * LDW;
    const int c = lane * 2;
    float acc0 = 0.0f, acc1 = 0.0f;
#pragma unroll
    for (int i = 0; i < 3; ++i) {
#pragma unroll
        for (int j = 0; j < 3; ++j) {
            const float bk = b0[i] * b1[j];
            const float2 t2 = *(const float2*)(Trow + (i * 3 + j) * COUT + c);
            acc0 = fmaf(bk, t2.x, acc0);
            acc1 = fmaf(bk, t2.y, acc1);
        }
    }
    float* o = out + (size_t)d * COUT + c;
    atomicAdd(o + 0, acc0);
    atomicAdd(o + 1, acc1);
}

__global__ __launch_bounds__(256) void relu_kernel(float* __restrict__ p, int n) {
    int t = blockIdx.x * 256 + threadIdx.x;
    if (t < n) p[t] = fmaxf(p[t], 0.0f);
}

extern "C" void kernel_launch(void* const* d_in, const int* in_sizes, int n_in,
                              void* d_out, int out_size, void* d_ws, size_t ws_size,
                              hipStream_t stream) {
    const float* x      = (const float*)d_in[0];
    const int*   ei     = (const int*)d_in[1];     // [2, NEDGES] row-major
    const float* pseudo = (const float*)d_in[2];
    const float* W1     = (const float*)d_in[3];
    const float* root1  = (const float*)d_in[4];
    const float* b1     = (const float*)d_in[5];
    const float* W2     = (const float*)d_in[6];
    const float* root2  = (const float*)d_in[7];
    const float* b2     = (const float*)d_in[8];
    float* out = (float*)d_out;

    float* T  = (float*)d_ws;                         // [NNODES, 640]
    float* h  = T + (size_t)NNODES * LDW;             // [NNODES, 64]
    float* Wf = h + (size_t)NNODES * COUT;            // [64, 640]

    const int* srcp = ei;
    const int* dstp = ei + NEDGES;

    const dim3 gemmGrid(NNODES / 16, LDW / 128);      // 6250 x 5 (exact)
    const int elemBlocks = (NNODES * COUT + 255) / 256;
    const int wfBlocks   = (CIN * LDW + 255) / 256;
    const int edgeBlocks = (NEDGES + 7) / 8;

    // ---- Layer 1: h = relu(spline_conv(x)) ----
    repack_kernel<<<wfBlocks, 256, 0, stream>>>(W1, root1, Wf);
    spline_gemm_kernel<<<gemmGrid, 256, 0, stream>>>(x, Wf, T, NNODES);
    init_out_kernel<<<elemBlocks, 256, 0, stream>>>(T, b1, h);
    edge_scatter_kernel<<<edgeBlocks, 256, 0, stream>>>(srcp, dstp, pseudo, T, h);
    relu_kernel<<<elemBlocks, 256, 0, stream>>>(h, NNODES * COUT);

    // ---- Layer 2: out = relu(spline_conv(h)) ----
    repack_kernel<<<wfBlocks, 256, 0, stream>>>(W2, root2, Wf);
    spline_gemm_kernel<<<gemmGrid, 256, 0, stream>>>(h, Wf, T, NNODES);
    init_out_kernel<<<elemBlocks, 256, 0, stream>>>(T, b2, out);
    edge_scatter_kernel<<<edgeBlocks, 256, 0, stream>>>(srcp, dstp, pseudo, T, out);
    relu_kernel<<<elemBlocks, 256, 0, stream>>>(out, NNODES * COUT);
}